// RNNModel_3762391351438
// MI455X (gfx1250) — compile-verified
//
#include <hip/hip_runtime.h>

typedef __attribute__((ext_vector_type(2))) float v2f;
typedef __attribute__((ext_vector_type(8))) float v8f;

// gfx1250 has V_TANH_F32 (TRANS op). Guard builtin name; fall back to
// tanh(x) = 1 - 2/(2^(2*log2e*x)+1)  (v_exp_f32 + v_rcp_f32, both TRANS).
__device__ __forceinline__ float fast_tanh(float x) {
#if __has_builtin(__builtin_amdgcn_tanhf)
  return __builtin_amdgcn_tanhf(x);
#elif __has_builtin(__builtin_amdgcn_tanh_f32)
  return __builtin_amdgcn_tanh_f32(x);
#else
  float e = __builtin_amdgcn_exp2f(x * 2.8853900817779268f); // 2*log2(e)
  return 1.0f - 2.0f * __builtin_amdgcn_rcpf(e + 1.0f);
#endif
}

// A-matrix row relabeling: rows 0..7 = W_hh rows 0..7; padding rows
// 8,9,12,13 duplicate W_hh rows 2,3,6,7; rows 10,11,14,15 are zero.
// This makes the WMMA D layout {row m | row m+8} coincide with the
// next step's B-operand layout {K | K+2}: no per-step lane shuffles.
__device__ __forceinline__ int rowmap(int M) {
  if (M < 8)   return M;
  if (M == 8)  return 2;
  if (M == 9)  return 3;
  if (M == 12) return 6;
  if (M == 13) return 7;
  return -1;
}

__global__ __launch_bounds__(32)
void rnn_scan_wmma(const float* __restrict__ x,     // [B, T]
                   const float* __restrict__ W_ih,  // [8,1]
                   const float* __restrict__ W_hh,  // [8,8]
                   const float* __restrict__ b_ih,  // [8]
                   const float* __restrict__ b_hh,  // [8]
                   const float* __restrict__ fc_w,  // [1,8]
                   const float* __restrict__ fc_b,  // [1]
                   float* __restrict__ out,         // [B,1]
                   int T) {
  const int  lane = threadIdx.x;        // one wave32 per block
  const int  col  = lane & 15;          // batch column inside the 16-wide tile
  const bool hi   = lane >= 16;
  const int  b    = blockIdx.x * 16 + col;

  // ---- constant A operands: A layout is lanes=M(0..15 twice), V0={K0|K2}, V1={K1|K3}
  const int rr = rowmap(col);
  const int k0 = hi ? 2 : 0;
  v2f A1, A2;                            // W_hh[:,0:4] and W_hh[:,4:8] (relabeled rows)
  if (rr >= 0) {
    A1.x = W_hh[rr * 8 + k0];
    A1.y = W_hh[rr * 8 + k0 + 1];
    A2.x = W_hh[rr * 8 + 4 + k0];
    A2.y = W_hh[rr * 8 + 4 + k0 + 1];
  } else {
    A1.x = 0.f; A1.y = 0.f; A2.x = 0.f; A2.y = 0.f;
  }

  // ---- per-lane xin constants: D row (hi ? m+8 : m) corresponds to hidden j
  float wih[8], bias[8];
#pragma unroll
  for (int m = 0; m < 8; ++m) {
    int j = hi ? rowmap(m + 8) : m;
    if (j < 0) j = 0;                    // unused padding rows: any finite value
    wih[m]  = W_ih[j];
    bias[m] = b_ih[j] + b_hh[j];
  }

  const float* xp = x + (size_t)b * (size_t)T;   // lanes n and n+16 share addresses

  // recurrence state: V0={h0|h2}, V1={h1|h3}, V4={h4|h6}, V5={h5|h7}; h(0)=0
  float H0 = 0.f, H1 = 0.f, H4 = 0.f, H5 = 0.f;

  for (int t = 0; t < T; t += 8) {
    __builtin_prefetch(xp + t + 96, 0, 3);       // global_prefetch_b8: stream x ahead
#pragma unroll
    for (int u = 0; u < 8; ++u) {
      const float xv = xp[t + u];
      v8f C;                                     // C = xin (independent of h)
#pragma unroll
      for (int m = 0; m < 8; ++m)
        C[m] = __builtin_fmaf(xv, wih[m], bias[m]);

      v2f Blo; Blo.x = H0; Blo.y = H1;           // hT rows 0..3  = {h0|h2},{h1|h3}
      v2f Bhi; Bhi.x = H4; Bhi.y = H5;           // hT rows 4..7  = {h4|h6},{h5|h7}

      v8f D = __builtin_amdgcn_wmma_f32_16x16x4_f32(false, A1, false, Blo,
                                                    (short)0, C, false, false);
      D     = __builtin_amdgcn_wmma_f32_16x16x4_f32(false, A2, false, Bhi,
                                                    (short)0, D, false, false);
      // rows {0|8},{1|9},{4|12},{5|13} carry all 8 hidden values; rest is padding
      H0 = fast_tanh(D[0]);
      H1 = fast_tanh(D[1]);
      H4 = fast_tanh(D[4]);
      H5 = fast_tanh(D[5]);
    }
  }

  // out[b] = fc_b + sum_j fc_w[j]*h[j]; low half holds j={0,1,4,5}, high j={2,3,6,7}
  const float f0 = hi ? fc_w[2] : fc_w[0];
  const float f1 = hi ? fc_w[3] : fc_w[1];
  const float f4 = hi ? fc_w[6] : fc_w[4];
  const float f5 = hi ? fc_w[7] : fc_w[5];
  float part = f0 * H0 + f1 * H1 + f4 * H4 + f5 * H5;
  part += __shfl_xor(part, 16);
  if (!hi) out[b] = part + fc_b[0];
}

extern "C" void kernel_launch(void* const* d_in, const int* in_sizes, int n_in,
                              void* d_out, int out_size, void* d_ws, size_t ws_size,
                              hipStream_t stream) {
  const float* x    = (const float*)d_in[0];
  const float* W_ih = (const float*)d_in[1];
  const float* W_hh = (const float*)d_in[2];
  const float* b_ih = (const float*)d_in[3];
  const float* b_hh = (const float*)d_in[4];
  const float* fc_w = (const float*)d_in[5];
  const float* fc_b = (const float*)d_in[6];
  float* out = (float*)d_out;

  const int B = out_size;            // 8192
  const int T = in_sizes[0] / B;     // 4096

  // 512 single-wave workgroups -> maximal spread over WGPs/SIMDs
  rnn_scan_wmma<<<dim3(B / 16), dim3(32), 0, stream>>>(
      x, W_ih, W_hh, b_ih, b_hh, fc_w, fc_b, out, T);
}